// STVQVae_42923903156868
// MI455X (gfx1250) — compile-verified
//
#include <hip/hip_runtime.h>
#include <stdint.h>

typedef float v2f __attribute__((ext_vector_type(2)));
typedef float v8f __attribute__((ext_vector_type(8)));
typedef int   gv4i __attribute__((vector_size(16)));   // matches builtin's V4i

#define DMODEL 768
#define LATENT 256
#define CBSZ   8192
#define MROWS  16384
#define LN_EPS 1e-5f

#define ROWS_PER_BLK 32          // 2 row tiles of 16
#define CB_STEP      64          // codebook rows staged per macro step (64 KB)
#define NSTEPS       (CBSZ / CB_STEP)   // 128

// padded pitches (floats): != 0 mod 64 banks, rows stay 16B aligned
#define EPITCH 260               // z_e rows      (260*4 = 1040 B)
#define ZPITCH 772               // patch rows    (772*4 = 3088 B)
#define BPITCH 260               // staged codebook rows (TDM pad gives this too)

// dynamic LDS layout (float indices)
#define OFF_E 0                                   // z_e   [32][EPITCH]
#define OFF_Z (OFF_E + ROWS_PER_BLK * EPITCH)     // zrow  [16][ZPITCH]
#define OFF_B (OFF_Z + 16 * ZPITCH)               // stage 2 x [64][BPITCH]
#define SMEM_FLOATS (OFF_B + 2 * CB_STEP * BPITCH)
#define SMEM_BYTES  (SMEM_FLOATS * 4)             // ~211 KB (<= 320 KB/WG)

#if __has_builtin(__builtin_amdgcn_tensor_load_to_lds) && \
    __has_builtin(__builtin_amdgcn_s_wait_tensorcnt)
#define HAVE_TDM 1
#else
#define HAVE_TDM 0
#endif

// ---------------------------------------------------------------------------
// CDNA5 async global->LDS helpers (fallback path when TDM builtin is absent)
// ---------------------------------------------------------------------------
__device__ __forceinline__ void async_b128(const float* gsrc, const float* ldst) {
#if __has_builtin(__builtin_amdgcn_global_load_async_to_lds_b128)
  __builtin_amdgcn_global_load_async_to_lds_b128(
      (__attribute__((address_space(1))) gv4i*)(uintptr_t)gsrc,
      (__attribute__((address_space(3))) gv4i*)(uintptr_t)(uint32_t)(uintptr_t)ldst,
      0, 0);
#else
  uint32_t loff = (uint32_t)(uintptr_t)ldst;   // low 32 bits == LDS byte offset
  asm volatile("global_load_async_to_lds_b128 %0, %1, off"
               :: "v"(loff), "v"(gsrc) : "memory");
#endif
}

__device__ __forceinline__ void wait_async0() {
#if __has_builtin(__builtin_amdgcn_s_wait_asynccnt)
  __builtin_amdgcn_s_wait_asynccnt(0);
#else
  asm volatile("s_wait_asynccnt 0x0" ::: "memory");
#endif
}

// stage 64 codebook rows (64 KB) of step s into LDS buffer (s&1): 16 b128/thread
__device__ __forceinline__ void stage_step(const float* __restrict__ codebook,
                                           float* smem, int s, int tid) {
  const float* gsrc = codebook + (size_t)s * (CB_STEP * LATENT);
  float* dst = smem + OFF_B + (s & 1) * (CB_STEP * BPITCH);
  #pragma unroll
  for (int i = 0; i < 16; ++i) {
    int o    = i * 256 + tid;        // 0..4095 : 16-byte chunk id
    int row  = o >> 6;               // 64 chunks per 1 KB row
    int chnk = o & 63;
    async_b128(gsrc + row * LATENT + chnk * 4,
               dst  + row * BPITCH + chnk * 4);
  }
}

#if HAVE_TDM
// ---------------------------------------------------------------------------
// Tensor Data Mover: one DMA op stages the whole 64x256 f32 tile, with
// hardware LDS padding of 4 DWORDs per 256 DWORDs (=> BPITCH 260 floats).
// ---------------------------------------------------------------------------
typedef unsigned int u32x4 __attribute__((vector_size(16)));
typedef int          i32x4 __attribute__((vector_size(16)));
typedef int          i32x8 __attribute__((vector_size(32)));

__device__ __forceinline__ void tdm_load_tile(const float* gsrc, const float* ldst) {
  uint64_t ga = (uint64_t)(uintptr_t)gsrc;
  uint32_t la = (uint32_t)(uintptr_t)ldst;     // LDS byte offset
  u32x4 g0;
  g0[0] = 1u;                                            // count=1 (valid), user D#
  g0[1] = la;                                            // lds_addr
  g0[2] = (uint32_t)(ga & 0xFFFFFFFFu);                  // global_addr[31:0]
  g0[3] = (uint32_t)((ga >> 32) & 0x01FFFFFFu) | (2u << 30);  // addr[56:32] | type=2
  i32x8 g1;
  g1[0] = (2 << 16)      // data_size = 4 bytes
        | (1 << 20)      // pad_enable
        | (7 << 22)      // pad_interval: 256 DWORDs
        | (3 << 25);     // pad_amount:   4 DWORDs
  g1[1] = (int)(256u << 16);                 // tensor_dim0[15:0] (bits 63:48)
  g1[2] = (int)(8192u << 16);                // dim0[31:16]=0 | tensor_dim1[15:0]
  g1[3] = (int)(256u << 16);                 // dim1[31:16]=0 | tile_dim0 = 256
  g1[4] = 64;                                // tile_dim1 = 64 | tile_dim2 = 0
  g1[5] = 256;                               // tensor_dim0_stride[31:0]
  g1[6] = 0;
  g1[7] = 0;
  i32x4 gz = {0, 0, 0, 0};                   // groups 2/3 unused (2-D tensor)
#if __clang_major__ >= 23
  i32x8 gz8 = {0, 0, 0, 0, 0, 0, 0, 0};
  __builtin_amdgcn_tensor_load_to_lds(g0, g1, gz, gz, gz8, 0);
#else
  __builtin_amdgcn_tensor_load_to_lds(g0, g1, gz, gz, 0);
#endif
}
#endif  // HAVE_TDM

// ---------------------------------------------------------------------------
// Kernel 1: negHalfNrm[c] = -0.5 * ||codebook[c]||^2
// ---------------------------------------------------------------------------
__global__ __launch_bounds__(256, 1)
void stvq_cbnorm(const float* __restrict__ cb, float* __restrict__ nrm) {
  const int row  = blockIdx.x * 8 + (threadIdx.x >> 5);
  const int lane = threadIdx.x & 31;
  const float* p = cb + (size_t)row * LATENT;
  float s = 0.f;
  #pragma unroll
  for (int j = 0; j < LATENT / 32; ++j) {
    float v = p[lane + 32 * j];
    s += v * v;
  }
  #pragma unroll
  for (int off = 16; off > 0; off >>= 1) s += __shfl_xor(s, off, 32);
  if (lane == 0) nrm[row] = -0.5f * s;
}

// ---------------------------------------------------------------------------
// Kernel 2: fused patchify + pos + LayerNorm + projection (WMMA f32) +
//           VQ argmax with TDM/async double-buffered codebook staging + gather.
// One 256-thread block (8 wave32) handles 32 rows; grid = 512.
// ---------------------------------------------------------------------------
__global__ __launch_bounds__(256, 1)
void stvq_fused(const float* __restrict__ x,
                const float* __restrict__ pos_space,
                const float* __restrict__ pos_time,
                const float* __restrict__ gamma,
                const float* __restrict__ beta,
                const float* __restrict__ w_proj,
                const float* __restrict__ b_proj,
                const float* __restrict__ codebook,
                const float* __restrict__ negHalfNrm,
                float* __restrict__ out) {
  extern __shared__ float smem[];
  __shared__ float sScore[4][ROWS_PER_BLK];
  __shared__ int   sIdx[4][ROWS_PER_BLK];
  __shared__ int   sFin[ROWS_PER_BLK];

  const int tid  = threadIdx.x;
  const int lane = tid & 31;
  const int wv   = tid >> 5;
  const int lc   = lane & 15;
  const int kh   = lane >> 4;
  const int base = blockIdx.x * ROWS_PER_BLK;

  float* sE = smem + OFF_E;
  float* sZ = smem + OFF_Z;

  // Kick off staging of codebook step 0 — overlaps with prep phases.
#if HAVE_TDM
  if (wv == 0) tdm_load_tile(codebook, smem + OFF_B);
#else
  stage_step(codebook, smem, 0, tid);
#endif

  // ===== Prep: two sequential 16-row tiles ================================
  for (int rt = 0; rt < 2; ++rt) {
    // -- patchify + positional embeddings -> sZ[16][ZPITCH] --
    for (int e = tid; e < 16 * DMODEL; e += 256) {
      int r  = e / DMODEL;
      int d  = e - r * DMODEL;
      int m  = base + rt * 16 + r;
      int b  = m >> 12;
      int f  = (m >> 8) & 15;
      int p  = m & 255;
      int ph = p >> 4, pw = p & 15;
      int c  = d >> 8;
      int rm = d & 255;
      int py = rm >> 4, px = rm & 15;
      int src = (((b * 3 + c) * 16 + f) << 16) + ((ph * 16 + py) << 8) + (pw * 16 + px);
      sZ[r * ZPITCH + d] = x[src] + pos_space[p * DMODEL + d] + pos_time[f * DMODEL + d];
    }
    __syncthreads();

    // -- LayerNorm: wave wv owns local rows 2wv, 2wv+1 --
    #pragma unroll
    for (int rr = 0; rr < 2; ++rr) {
      float* zr = sZ + (2 * wv + rr) * ZPITCH;
      float s = 0.f;
      #pragma unroll
      for (int j = 0; j < DMODEL / 32; ++j) s += zr[lane + 32 * j];
      #pragma unroll
      for (int off = 16; off > 0; off >>= 1) s += __shfl_xor(s, off, 32);
      float mu = s * (1.0f / DMODEL);
      float v = 0.f;
      #pragma unroll
      for (int j = 0; j < DMODEL / 32; ++j) {
        float d0 = zr[lane + 32 * j] - mu;
        v += d0 * d0;
      }
      #pragma unroll
      for (int off = 16; off > 0; off >>= 1) v += __shfl_xor(v, off, 32);
      float rstd = rsqrtf(v * (1.0f / DMODEL) + LN_EPS);
      #pragma unroll
      for (int j = 0; j < DMODEL / 32; ++j) {
        int k = lane + 32 * j;
        zr[k] = (zr[k] - mu) * rstd * gamma[k] + beta[k];
      }
    }
    __syncthreads();

    // -- projection z_e = z @ w_proj + b (WMMA f32 16x16x4) --
    v8f accT[2];
    #pragma unroll
    for (int tt = 0; tt < 2; ++tt) {
      const int n0 = 16 * (2 * wv + tt);
      v8f acc = {};
      #pragma unroll 4
      for (int k = 0; k < DMODEL; k += 4) {
        int kb = k + 2 * kh;
        v2f a = *(const v2f*)(sZ + lc * ZPITCH + kb);
        v2f bf;
        bf.x = w_proj[kb * LATENT + n0 + lc];
        bf.y = w_proj[(kb + 1) * LATENT + n0 + lc];
        acc = __builtin_amdgcn_wmma_f32_16x16x4_f32(false, a, false, bf,
                                                    (short)0, acc, false, false);
      }
      float bias = b_proj[n0 + lc];
      #pragma unroll
      for (int i = 0; i < 8; ++i) acc[i] += bias;
      accT[tt] = acc;
    }
    __syncthreads();   // all reads of sZ complete before next tile reuses it

    #pragma unroll
    for (int tt = 0; tt < 2; ++tt) {
      const int n0 = 16 * (2 * wv + tt);
      #pragma unroll
      for (int i = 0; i < 8; ++i) {
        int row = rt * 16 + i + 8 * kh;
        sE[row * EPITCH + n0 + lc] = accT[tt][i];
      }
    }
  }
  __syncthreads();   // z_e[32][..] complete

  // ===== VQ: argmax_c ( e.c - 0.5||c||^2 ), double-buffered staging =======
  const int rg = wv >> 2;
  const int cg = wv & 3;
  const float* eBase = sE + (rg * 16 + lc) * EPITCH + 2 * kh;

  float best[8];
  int   bidx[8];
  #pragma unroll
  for (int i = 0; i < 8; ++i) { best[i] = -3.402823466e38f; bidx[i] = 0; }

  for (int s = 0; s < NSTEPS; ++s) {
#if HAVE_TDM
    if (wv == 0) __builtin_amdgcn_s_wait_tensorcnt(0);  // tile (s&1) landed
    __syncthreads();   // release everyone; nobody still computing (s-1)
    if (s + 1 < NSTEPS && wv == 0)
      tdm_load_tile(codebook + (size_t)(s + 1) * (CB_STEP * LATENT),
                    smem + OFF_B + ((s + 1) & 1) * (CB_STEP * BPITCH));
#else
    wait_async0();
    __syncthreads();
    if (s + 1 < NSTEPS) stage_step(codebook, smem, s + 1, tid);
#endif

    const float* bBase = smem + OFF_B + (s & 1) * (CB_STEP * BPITCH)
                       + (16 * cg + lc) * BPITCH + 2 * kh;
    v8f acc = {};
    #pragma unroll
    for (int kk = 0; kk < LATENT / 4; ++kk) {
      v2f a = *(const v2f*)(eBase + 4 * kk);
      v2f bf = *(const v2f*)(bBase + 4 * kk);
      acc = __builtin_amdgcn_wmma_f32_16x16x4_f32(false, a, false, bf,
                                                  (short)0, acc, false, false);
    }
    const int   n  = s * CB_STEP + 16 * cg + lc;
    const float ns = negHalfNrm[n];
    #pragma unroll
    for (int i = 0; i < 8; ++i) {
      float sc = acc[i] + ns;
      if (sc > best[i]) { best[i] = sc; bidx[i] = n; }
    }
  }

  // reduce across the 16 lanes sharing each row
  #pragma unroll
  for (int off = 8; off > 0; off >>= 1) {
    #pragma unroll
    for (int i = 0; i < 8; ++i) {
      float os = __shfl_xor(best[i], off, 32);
      int   oi = __shfl_xor(bidx[i], off, 32);
      if (os > best[i] || (os == best[i] && oi < bidx[i])) { best[i] = os; bidx[i] = oi; }
    }
  }
  if (lc == 0) {
    #pragma unroll
    for (int i = 0; i < 8; ++i) {
      int row = rg * 16 + i + 8 * kh;
      sScore[cg][row] = best[i];
      sIdx[cg][row]   = bidx[i];
    }
  }
  __syncthreads();

  // final cross-colgroup reduce: one thread per row
  if (tid < ROWS_PER_BLK) {
    float bs = sScore[0][tid];
    int   bi = sIdx[0][tid];
    #pragma unroll
    for (int c2 = 1; c2 < 4; ++c2) {
      float s2 = sScore[c2][tid];
      int   i2 = sIdx[c2][tid];
      if (s2 > bs || (s2 == bs && i2 < bi)) { bs = s2; bi = i2; }
    }
    sFin[tid] = bi;
  }
  __syncthreads();

  // gather: z_q[row] = codebook[idx[row]]
  for (int e = tid; e < ROWS_PER_BLK * LATENT; e += 256) {
    int r = e >> 8;
    int c = e & 255;
    out[(size_t)(base + r) * LATENT + c] = codebook[(size_t)sFin[r] * LATENT + c];
  }
}

// ---------------------------------------------------------------------------
extern "C" void kernel_launch(void* const* d_in, const int* in_sizes, int n_in,
                              void* d_out, int out_size, void* d_ws, size_t ws_size,
                              hipStream_t stream) {
  const float* x         = (const float*)d_in[0];
  const float* pos_space = (const float*)d_in[1];
  const float* pos_time  = (const float*)d_in[2];
  const float* ln_gamma  = (const float*)d_in[3];
  const float* ln_beta   = (const float*)d_in[4];
  const float* w_proj    = (const float*)d_in[5];
  const float* b_proj    = (const float*)d_in[6];
  const float* codebook  = (const float*)d_in[7];

  float* nrm = (float*)d_ws;  // 8192 floats = 32 KB scratch

  // allow > 64 KB dynamic LDS (CDNA5: up to 320 KB per workgroup)
  (void)hipFuncSetAttribute((const void*)stvq_fused,
                            hipFuncAttributeMaxDynamicSharedMemorySize,
                            SMEM_BYTES);

  stvq_cbnorm<<<CBSZ / 8, 256, 0, stream>>>(codebook, nrm);
  stvq_fused<<<MROWS / ROWS_PER_BLK, 256, SMEM_BYTES, stream>>>(
      x, pos_space, pos_time, ln_gamma, ln_beta,
      w_proj, b_proj, codebook, nrm, (float*)d_out);
}